// GATConv_79139067396226
// MI455X (gfx1250) — compile-verified
//
#include <hip/hip_runtime.h>
#include <math.h>

// ---------------------------------------------------------------------------
// GATConv forward for MI455X (gfx1250, wave32).
//   h = x @ W           -> WMMA f32 16x16x4 (full f32 precision)
//   edge softmax/scatter-> L2-resident (192MB) gather + atomic pipeline
// Workspace layout (floats): h[N*128] | a_src[N*4] | a_dst[N*4] | m[N*4] | denom[N*4]
//   total = N*144*4 bytes = 57.6 MB for N=100000.
// d_out ([N,32] f32) is used directly as the scatter accumulator, finalized in place.
// ---------------------------------------------------------------------------

typedef __attribute__((ext_vector_type(2))) float v2f;
typedef __attribute__((ext_vector_type(8))) float v8f;

#define HEADS 4
#define CDIM 32
#define FDIM 128
#define NEG_SLOPE 0.2f

__device__ __forceinline__ float leaky(float v) { return v > 0.f ? v : NEG_SLOPE * v; }

// float atomic max via signed/unsigned int ordering trick (valid incl. -inf init)
__device__ __forceinline__ void atomicMaxFloat(float* addr, float val) {
  if (val >= 0.f)
    atomicMax((int*)addr, __float_as_int(val));
  else
    atomicMin((unsigned int*)addr, __float_as_uint(val));
}

// wave-uniform predicate -> force scalar branch (SCC) instead of exec-mask path
__device__ __forceinline__ bool uniform_bool(bool p) {
  return __builtin_amdgcn_readfirstlane(p ? 1 : 0) != 0;
}

// ---------------------------------------------------------------------------
// Kernel 1: h = x @ W  via V_WMMA_F32_16X16X4_F32.
// One wave per 16x16 output tile; 8 col-tiles share a row-tile within a block
// so the A (x) rows hit in L0. K loop: 128/4 = 32 WMMA issues per tile.
// Epilogue: scalar-branched full-tile fast path -> 8-store clause off one base.
// ---------------------------------------------------------------------------
__global__ void gat_gemm_wmma(const float* __restrict__ x, const float* __restrict__ W,
                              float* __restrict__ h, int n) {
  const int wave = blockIdx.x * (blockDim.x >> 5) + (threadIdx.x >> 5);
  const int lane = threadIdx.x & 31;
  const int row_tile = wave >> 3;          // 8 column tiles of 16 (128 cols)
  const int col_tile = wave & 7;
  const int row0 = row_tile * 16;
  if (uniform_bool(row0 >= n)) return;     // scalar branch, EXEC stays all-ones
  const int col0 = col_tile * 16;

  const bool full_tile = uniform_bool(row0 + 16 <= n);

  const int lhalf = lane >> 4;             // 0: K-pair {0,1} / rows 0-7 on store
  const int lmod  = lane & 15;

  int arow = row0 + lmod;                  // A-matrix row this lane feeds
  if (!full_tile && arow >= n) arow = n - 1;  // clamp only in tail tile
  const float* __restrict__ xrow = x + (size_t)arow * FDIM;

  v8f acc = {};
  for (int k0 = 0; k0 < FDIM; k0 += 4) {
    const int kb = k0 + 2 * lhalf;
    v2f a, b;
    a.x = xrow[kb + 0];                                  // A[M=lmod][K=kb]
    a.y = xrow[kb + 1];                                  // A[M=lmod][K=kb+1]
    b.x = W[(size_t)(kb + 0) * FDIM + col0 + lmod];      // B[K=kb  ][N=lmod]
    b.y = W[(size_t)(kb + 1) * FDIM + col0 + lmod];      // B[K=kb+1][N=lmod]
    acc = __builtin_amdgcn_wmma_f32_16x16x4_f32(
        /*neg_a=*/false, a, /*neg_b=*/false, b,
        /*c_mod=*/(short)0, acc, /*reuse_a=*/false, /*reuse_b=*/false);
  }

  // C/D layout: VGPR r -> row (row0 + 8*lhalf + r), col (col0 + lmod)
  const int rbase = row0 + 8 * lhalf;
  if (full_tile) {
    // scalar-uniform branch: 8 unconditional stores, one base + imm offsets
    float* __restrict__ hb = h + (size_t)rbase * FDIM + col0 + lmod;
#pragma unroll
    for (int r = 0; r < 8; ++r) hb[r * FDIM] = acc[r];
  } else {
#pragma unroll
    for (int r = 0; r < 8; ++r) {
      const int row = rbase + r;
      if (row < n) h[(size_t)row * FDIM + col0 + lmod] = acc[r];
    }
  }
}

// ---------------------------------------------------------------------------
// Kernel 2: init accumulators (m = -inf, denom = 0, out_acc = 0)
// ---------------------------------------------------------------------------
__global__ void gat_init(float* __restrict__ m, float* __restrict__ denom,
                         float* __restrict__ out_acc, int n) {
  const int i = blockIdx.x * blockDim.x + threadIdx.x;
  if (i < n * HEADS) { m[i] = -INFINITY; denom[i] = 0.f; }
  if (i < n * CDIM)  out_acc[i] = 0.f;
}

// ---------------------------------------------------------------------------
// Kernel 3: per-node attention scores a_src/a_dst [N,H]
// ---------------------------------------------------------------------------
__global__ void gat_scores(const float* __restrict__ h,
                           const float* __restrict__ att_src,
                           const float* __restrict__ att_dst,
                           float* __restrict__ a_src, float* __restrict__ a_dst, int n) {
  const int i = blockIdx.x * blockDim.x + threadIdx.x;   // i = node*H + head
  if (i >= n * HEADS) return;
  const int node = i >> 2, head = i & 3;
  const float* __restrict__ hp = h + (size_t)node * FDIM + head * CDIM;
  const float* __restrict__ as = att_src + head * CDIM;
  const float* __restrict__ ad = att_dst + head * CDIM;
  float s = 0.f, d = 0.f;
#pragma unroll 8
  for (int c = 0; c < CDIM; ++c) { const float v = hp[c]; s += v * as[c]; d += v * ad[c]; }
  a_src[i] = s;
  a_dst[i] = d;
}

// ---------------------------------------------------------------------------
// Kernel 4: segment max of leaky_relu logits over dst (thread per edge*head)
// Edges e >= E are the implicit self-loops (src = dst = e - E).
// ---------------------------------------------------------------------------
__global__ void gat_edge_max(const int* __restrict__ src, const int* __restrict__ dst,
                             const float* __restrict__ a_src, const float* __restrict__ a_dst,
                             float* __restrict__ m, int E, int EP) {
  const int i = blockIdx.x * blockDim.x + threadIdx.x;
  if (i >= EP * HEADS) return;
  const int e = i >> 2, head = i & 3;
  const int s = (e < E) ? src[e] : (e - E);
  const int d = (e < E) ? dst[e] : (e - E);
  const float v = leaky(a_src[s * HEADS + head] + a_dst[d * HEADS + head]);
  atomicMaxFloat(&m[d * HEADS + head], v);
}

// ---------------------------------------------------------------------------
// Kernel 5: softmax denominator (thread per edge*head)
// ---------------------------------------------------------------------------
__global__ void gat_edge_denom(const int* __restrict__ src, const int* __restrict__ dst,
                               const float* __restrict__ a_src, const float* __restrict__ a_dst,
                               const float* __restrict__ m, float* __restrict__ denom,
                               int E, int EP) {
  const int i = blockIdx.x * blockDim.x + threadIdx.x;
  if (i >= EP * HEADS) return;
  const int e = i >> 2, head = i & 3;
  const int s = (e < E) ? src[e] : (e - E);
  const int d = (e < E) ? dst[e] : (e - E);
  const float v = leaky(a_src[s * HEADS + head] + a_dst[d * HEADS + head]);
  float mm = m[d * HEADS + head];
  if (!isfinite(mm)) mm = 0.f;
  atomicAdd(&denom[d * HEADS + head], expf(v - mm));
}

// ---------------------------------------------------------------------------
// Kernel 6: weighted message scatter. ONE WAVE PER EDGE, lane = channel c.
// Heads are folded in registers -> one atomicAdd per (edge, c):
//   out_acc[d, c] += sum_h alpha[e,h] * h[s, h*32 + c]
// ---------------------------------------------------------------------------
__global__ void gat_scatter(const int* __restrict__ src, const int* __restrict__ dst,
                            const float* __restrict__ h,
                            const float* __restrict__ a_src, const float* __restrict__ a_dst,
                            const float* __restrict__ m, const float* __restrict__ denom,
                            float* __restrict__ out_acc, int E, int EP) {
  const int e = blockIdx.x * (blockDim.x >> 5) + (threadIdx.x >> 5);
  const int lane = threadIdx.x & 31;       // channel within head
  if (uniform_bool(e >= EP)) return;       // uniform per wave
  const int s = (e < E) ? src[e] : (e - E);
  const int d = (e < E) ? dst[e] : (e - E);
  const float* __restrict__ hs = h + (size_t)s * FDIM;
  float val = 0.f;
#pragma unroll
  for (int head = 0; head < HEADS; ++head) {
    const float lg = leaky(a_src[s * HEADS + head] + a_dst[d * HEADS + head]);
    float mm = m[d * HEADS + head];
    if (!isfinite(mm)) mm = 0.f;
    const float alpha = expf(lg - mm) / (denom[d * HEADS + head] + 1e-16f);
    val += alpha * hs[head * CDIM + lane];
  }
  atomicAdd(&out_acc[(size_t)d * CDIM + lane], val);
}

// ---------------------------------------------------------------------------
// Kernel 7: finalize in place: out = relu(acc / H + bias)
// ---------------------------------------------------------------------------
__global__ void gat_final(float* __restrict__ out, const float* __restrict__ bias, int n) {
  const int i = blockIdx.x * blockDim.x + threadIdx.x;
  if (i >= n * CDIM) return;
  const float v = out[i] * (1.0f / HEADS) + bias[i & (CDIM - 1)];
  out[i] = v > 0.f ? v : 0.f;
}

// ---------------------------------------------------------------------------
extern "C" void kernel_launch(void* const* d_in, const int* in_sizes, int n_in,
                              void* d_out, int out_size, void* d_ws, size_t ws_size,
                              hipStream_t stream) {
  const float* x       = (const float*)d_in[0];
  const int*   ei      = (const int*)d_in[1];
  const float* W       = (const float*)d_in[2];
  const float* att_src = (const float*)d_in[3];
  const float* att_dst = (const float*)d_in[4];
  const float* bias    = (const float*)d_in[5];
  float* out = (float*)d_out;

  const int n  = in_sizes[0] / FDIM;   // nodes
  const int E  = in_sizes[1] / 2;      // edges (before self loops)
  const int EP = E + n;                // edges incl. self loops

  float* ws    = (float*)d_ws;
  float* h     = ws;                               // n*128
  float* asrc  = h + (size_t)n * FDIM;             // n*4
  float* adst  = asrc + (size_t)n * HEADS;         // n*4
  float* m     = adst + (size_t)n * HEADS;         // n*4
  float* denom = m + (size_t)n * HEADS;            // n*4

  const int* src = ei;
  const int* dst = ei + E;

  // 1) GEMM (WMMA): one wave per 16x16 tile, 8 col tiles per 16-row strip
  {
    const int row_tiles = (n + 15) / 16;
    const int waves = row_tiles * (FDIM / 16);
    const int blocks = (waves + 7) / 8;            // 8 waves (256 thr) per block
    gat_gemm_wmma<<<blocks, 256, 0, stream>>>(x, W, h, n);
  }
  // 2) init m/denom/out accumulator
  gat_init<<<(n * CDIM + 255) / 256, 256, 0, stream>>>(m, denom, out, n);
  // 3) per-node scores
  gat_scores<<<(n * HEADS + 255) / 256, 256, 0, stream>>>(h, att_src, att_dst, asrc, adst, n);
  // 4) segment max
  gat_edge_max<<<((long long)EP * HEADS + 255) / 256, 256, 0, stream>>>(src, dst, asrc, adst, m, E, EP);
  // 5) softmax denominator
  gat_edge_denom<<<((long long)EP * HEADS + 255) / 256, 256, 0, stream>>>(src, dst, asrc, adst, m, denom, E, EP);
  // 6) message scatter: one wave per edge (8 edges / 256-thr block)
  gat_scatter<<<((long long)EP + 7) / 8, 256, 0, stream>>>(src, dst, h, asrc, adst, m, denom, out, E, EP);
  // 7) head mean + bias + relu, in place on d_out
  gat_final<<<(n * CDIM + 255) / 256, 256, 0, stream>>>(out, bias, n);
}